// GAT_58935541235964
// MI455X (gfx1250) — compile-verified
//
#include <hip/hip_runtime.h>
#include <hip/hip_bf16.h>

// Problem constants (match reference)
#define DFEAT   200
#define NNODES  100000
#define NEDGES  400000
#define NBATCH  4096
#define NLAYERS 2
#define LRELU_ALPHA 0.01f

#define TILESN  13          // ceil(200/16) column tiles
#define BROWS   256         // rows of C per block (8 waves x 2 row-tiles)
#define BSTRIDE 202         // LDS stride for transposed B tile (conflict-free, 8B-aligned)

typedef float v2f __attribute__((ext_vector_type(2)));
typedef float v8f __attribute__((ext_vector_type(8)));

__device__ __forceinline__ v8f wmma_f32(v2f a, v2f b, v8f c) {
    return __builtin_amdgcn_wmma_f32_16x16x4_f32(false, a, false, b,
                                                 (short)0, c, false, false);
}

// ---------------------------------------------------------------------------
// Order-preserving float<->uint encoding so segment-max can use atomicMax(u32)
// ---------------------------------------------------------------------------
__device__ __forceinline__ unsigned f2ord(float f) {
    unsigned u = __float_as_uint(f);
    return (u & 0x80000000u) ? ~u : (u | 0x80000000u);
}
__device__ __forceinline__ float ord2f(unsigned u) {
    u = (u & 0x80000000u) ? (u & 0x7FFFFFFFu) : ~u;
    return __uint_as_float(u);
}

// ---------------------------------------------------------------------------
// Init: zero agg accumulator, reset per-node max (encoded -inf = 0) and denom
// ---------------------------------------------------------------------------
__global__ __launch_bounds__(256) void k_init(float* __restrict__ agg,
                                              unsigned* __restrict__ menc,
                                              float* __restrict__ denom,
                                              int nNodes, int total) {
    int i = blockIdx.x * blockDim.x + threadIdx.x;
    if (i < total) agg[i] = 0.0f;
    if (i < nNodes) { menc[i] = 0u; denom[i] = 0.0f; }
}

// ---------------------------------------------------------------------------
// f32 WMMA GEMM: C[M x 200] = A[M x 200] * B[200 x 200].
// Block = 256 threads = 8 waves. Each block owns ONE 16-col tile of B
// (staged transposed in LDS, loaded once, reused by all waves) and 256 rows
// of A/C. Each wave computes a 32x16 C slab with two f32 accumulators so the
// LDS b-fragment is reused across two WMMAs per K-step.
// All loads are branchless (clamped addresses); only stores are masked.
// ---------------------------------------------------------------------------
__global__ __launch_bounds__(256) void k_gemm(const float* __restrict__ A,
                                              const float* __restrict__ Bw,
                                              float* __restrict__ C,
                                              int M) {
    __shared__ float Bs[16 * BSTRIDE];        // transposed B tile: Bs[col][k]

    const int tn = blockIdx.x % TILESN;       // column tile
    const int mb = blockIdx.x / TILESN;       // 256-row block

    // Cooperative, coalesced load of B[0:200, tn*16:+16] -> LDS (transposed).
    for (int e = threadIdx.x; e < DFEAT * 16; e += 256) {
        int k = e >> 4, c = e & 15;
        int gc = tn * 16 + c;
        Bs[c * BSTRIDE + k] = (gc < DFEAT) ? Bw[(size_t)k * DFEAT + gc] : 0.0f;
    }
    __syncthreads();

    const int lane = threadIdx.x & 31;
    const int wave = threadIdx.x >> 5;
    const int half = lane >> 4;               // selects K pair within step-of-4
    const int l16  = lane & 15;

    const int row0 = mb * BROWS + wave * 32 + l16;   // acc0 fragment row
    const int row1 = row0 + 16;                      // acc1 fragment row
    const int r0c  = min(row0, M - 1);               // clamped (edge block)
    const int r1c  = min(row1, M - 1);

    const float* pa0 = A + (size_t)r0c * DFEAT + half * 2;
    const float* pa1 = A + (size_t)r1c * DFEAT + half * 2;
    const float* pb  = Bs + l16 * BSTRIDE + half * 2;

    v8f acc0 = {}, acc1 = {};
#pragma unroll 5
    for (int k0 = 0; k0 < DFEAT; k0 += 4) {
        v2f a0 = *(const v2f*)(pa0 + k0);     // global_load_b64
        v2f a1 = *(const v2f*)(pa1 + k0);     // global_load_b64
        v2f b  = *(const v2f*)(pb + k0);      // ds_load_b64
        acc0 = wmma_f32(a0, b, acc0);
        acc1 = wmma_f32(a1, b, acc1);
    }

    const int col = tn * 16 + l16;
    if (col < DFEAT) {
        const int rb0 = mb * BROWS + wave * 32 + half * 8;
#pragma unroll
        for (int r = 0; r < 8; ++r) {
            int ra = rb0 + r, rb = ra + 16;
            if (ra < M) C[(size_t)ra * DFEAT + col] = acc0[r];
            if (rb < M) C[(size_t)rb * DFEAT + col] = acc1[r];
        }
    }
}

// ---------------------------------------------------------------------------
// Highway GEMM fused: Xnew = sigma * sigmoid(agg) + (1-sigma) * X,
// sigma = sigmoid(X @ W_hw + b_hw). Same WMMA tiling as k_gemm.
// ---------------------------------------------------------------------------
__global__ __launch_bounds__(256) void k_highway(const float* __restrict__ X,
                                                 const float* __restrict__ Whw,
                                                 const float* __restrict__ bhw,
                                                 const float* __restrict__ agg,
                                                 float* __restrict__ Xnew,
                                                 int M) {
    __shared__ float Bs[16 * BSTRIDE];

    const int tn = blockIdx.x % TILESN;
    const int mb = blockIdx.x / TILESN;

    for (int e = threadIdx.x; e < DFEAT * 16; e += 256) {
        int k = e >> 4, c = e & 15;
        int gc = tn * 16 + c;
        Bs[c * BSTRIDE + k] = (gc < DFEAT) ? Whw[(size_t)k * DFEAT + gc] : 0.0f;
    }
    __syncthreads();

    const int lane = threadIdx.x & 31;
    const int wave = threadIdx.x >> 5;
    const int half = lane >> 4;
    const int l16  = lane & 15;

    const int row0 = mb * BROWS + wave * 32 + l16;
    const int row1 = row0 + 16;
    const int r0c  = min(row0, M - 1);
    const int r1c  = min(row1, M - 1);

    const float* pa0 = X + (size_t)r0c * DFEAT + half * 2;
    const float* pa1 = X + (size_t)r1c * DFEAT + half * 2;
    const float* pb  = Bs + l16 * BSTRIDE + half * 2;

    v8f acc0 = {}, acc1 = {};
#pragma unroll 5
    for (int k0 = 0; k0 < DFEAT; k0 += 4) {
        v2f a0 = *(const v2f*)(pa0 + k0);
        v2f a1 = *(const v2f*)(pa1 + k0);
        v2f b  = *(const v2f*)(pb + k0);
        acc0 = wmma_f32(a0, b, acc0);
        acc1 = wmma_f32(a1, b, acc1);
    }

    const int col = tn * 16 + l16;
    if (col < DFEAT) {
        const float bc = bhw[col];
        const int rb0 = mb * BROWS + wave * 32 + half * 8;
#pragma unroll
        for (int r = 0; r < 8; ++r) {
            int ra = rb0 + r, rb = ra + 16;
            if (ra < M) {
                size_t idx = (size_t)ra * DFEAT + col;
                float sigma = 1.0f / (1.0f + expf(-(acc0[r] + bc)));
                float gat   = 1.0f / (1.0f + expf(-agg[idx]));     // sigmoid(out)
                Xnew[idx] = sigma * gat + (1.0f - sigma) * X[idx];
            }
            if (rb < M) {
                size_t idx = (size_t)rb * DFEAT + col;
                float sigma = 1.0f / (1.0f + expf(-(acc1[r] + bc)));
                float gat   = 1.0f / (1.0f + expf(-agg[idx]));
                Xnew[idx] = sigma * gat + (1.0f - sigma) * X[idx];
            }
        }
    }
}

// ---------------------------------------------------------------------------
// Per-node attention scores: wave-per-node dot products with a[:D], a[D:2D]
// ---------------------------------------------------------------------------
__global__ __launch_bounds__(256) void k_scores(const float* __restrict__ Wh,
                                                const float* __restrict__ att,
                                                float* __restrict__ ssrc,
                                                float* __restrict__ sdst,
                                                int M) {
    int node = (int)((blockIdx.x * (size_t)blockDim.x + threadIdx.x) >> 5);
    int lane = threadIdx.x & 31;
    if (node >= M) return;
    const float* row = Wh + (size_t)node * DFEAT;
    float s1 = 0.0f, s2 = 0.0f;
    for (int d = lane; d < DFEAT; d += 32) {
        float v = row[d];
        s1 += v * att[d];
        s2 += v * att[DFEAT + d];
    }
#pragma unroll
    for (int off = 16; off > 0; off >>= 1) {
        s1 += __shfl_down(s1, off, 32);
        s2 += __shfl_down(s2, off, 32);
    }
    if (lane == 0) { ssrc[node] = s1; sdst[node] = s2; }
}

// ---------------------------------------------------------------------------
// Edge logits + segment max (encoded-u32 atomicMax)
// ---------------------------------------------------------------------------
__global__ __launch_bounds__(256) void k_edge_logits(const int* __restrict__ src,
                                                     const int* __restrict__ dst,
                                                     const float* __restrict__ ssrc,
                                                     const float* __restrict__ sdst,
                                                     float* __restrict__ ebuf,
                                                     unsigned* __restrict__ menc,
                                                     int E) {
    int i = blockIdx.x * blockDim.x + threadIdx.x;
    if (i >= E) return;
    float v = ssrc[src[i]] + sdst[dst[i]];
    v = (v > 0.0f) ? v : LRELU_ALPHA * v;     // leaky relu
    ebuf[i] = v;
    atomicMax(&menc[dst[i]], f2ord(v));
}

// ---------------------------------------------------------------------------
// exp(e - m[dst]) and segment denom
// ---------------------------------------------------------------------------
__global__ __launch_bounds__(256) void k_edge_exp(const int* __restrict__ dst,
                                                  float* __restrict__ ebuf,
                                                  const unsigned* __restrict__ menc,
                                                  float* __restrict__ denom,
                                                  int E) {
    int i = blockIdx.x * blockDim.x + threadIdx.x;
    if (i >= E) return;
    int d = dst[i];
    float ex = expf(ebuf[i] - ord2f(menc[d]));
    ebuf[i] = ex;
    atomicAdd(&denom[d], ex);
}

// ---------------------------------------------------------------------------
// Wave-per-edge aggregation: agg[dst] += att * Wh[src]
// ---------------------------------------------------------------------------
__global__ __launch_bounds__(256) void k_edge_agg(const int* __restrict__ src,
                                                  const int* __restrict__ dst,
                                                  const float* __restrict__ ebuf,
                                                  const float* __restrict__ denom,
                                                  const float* __restrict__ Wh,
                                                  float* __restrict__ agg,
                                                  int E) {
    int e    = (int)((blockIdx.x * (size_t)blockDim.x + threadIdx.x) >> 5);
    int lane = threadIdx.x & 31;
    if (e >= E) return;
    int s = src[e], t = dst[e];
    float att = ebuf[e] / fmaxf(denom[t], 1e-9f);
    const float* wr = Wh  + (size_t)s * DFEAT;
    float*       ar = agg + (size_t)t * DFEAT;
    for (int d = lane; d < DFEAT; d += 32)
        atomicAdd(&ar[d], att * wr[d]);
}

// ---------------------------------------------------------------------------
// Final gather: out = concat(x[batch_h], rel[batch_r], x[batch_t]) flat
// ---------------------------------------------------------------------------
__global__ __launch_bounds__(256) void k_gather(const float* __restrict__ x,
                                                const float* __restrict__ rel,
                                                const int* __restrict__ bh,
                                                const int* __restrict__ br,
                                                const int* __restrict__ bt,
                                                float* __restrict__ out,
                                                int B) {
    int i = blockIdx.x * blockDim.x + threadIdx.x;
    int total = B * DFEAT;
    if (i >= total) return;
    int b = i / DFEAT, d = i % DFEAT;
    out[i]             = x[(size_t)bh[b] * DFEAT + d];
    out[total + i]     = rel[(size_t)br[b] * DFEAT + d];
    out[2 * total + i] = x[(size_t)bt[b] * DFEAT + d];
}

// ---------------------------------------------------------------------------
extern "C" void kernel_launch(void* const* d_in, const int* in_sizes, int n_in,
                              void* d_out, int out_size, void* d_ws, size_t ws_size,
                              hipStream_t stream) {
    const float* ent  = (const float*)d_in[0];
    const float* rel  = (const float*)d_in[1];
    const float* Wg   = (const float*)d_in[2];
    const float* atta = (const float*)d_in[3];
    const float* Whw  = (const float*)d_in[4];
    const float* bhw  = (const float*)d_in[5];
    const int* esrc   = (const int*)d_in[6];
    const int* edst   = (const int*)d_in[7];
    const int* bh     = (const int*)d_in[8];
    const int* br     = (const int*)d_in[9];
    const int* bt     = (const int*)d_in[10];
    float* out = (float*)d_out;

    const size_t NM = (size_t)NNODES * DFEAT;
    float* P0       = (float*)d_ws;          // rotating x / Wh / x_next panels
    float* P1       = P0 + NM;
    float* AGG      = P1 + NM;               // attention aggregation accumulator
    float* ssrc     = AGG + NM;
    float* sdst     = ssrc + NNODES;
    unsigned* menc  = (unsigned*)(sdst + NNODES);
    float* denom    = (float*)(menc + NNODES);
    float* ebuf     = denom + NNODES;        // per-edge logit -> exp

    // x0 = ent_embed (don't mutate inputs)
    hipMemcpyAsync(P0, ent, NM * sizeof(float), hipMemcpyDeviceToDevice, stream);

    float* xcur = P0;
    float* xalt = P1;   // holds Wh during layer, then overwritten with x_next

    const int rowBlocks  = (NNODES + BROWS - 1) / BROWS;   // 391
    const int gemmBlocks = rowBlocks * TILESN;             // 5083
    const int initBlocks = (int)((NM + 255) / 256);
    const int edgeBlocks = (NEDGES + 255) / 256;

    for (int l = 0; l < NLAYERS; ++l) {
        k_init<<<initBlocks, 256, 0, stream>>>(AGG, menc, denom, NNODES, (int)NM);
        // Wh = x @ W_gat[l]   (WMMA f32, LDS-staged B tile)
        k_gemm<<<gemmBlocks, 256, 0, stream>>>(xcur, Wg + (size_t)l * DFEAT * DFEAT,
                                               xalt, NNODES);
        k_scores<<<NNODES / 8, 256, 0, stream>>>(xalt, atta + (size_t)l * 2 * DFEAT,
                                                 ssrc, sdst, NNODES);
        k_edge_logits<<<edgeBlocks, 256, 0, stream>>>(esrc, edst, ssrc, sdst,
                                                      ebuf, menc, NEDGES);
        k_edge_exp<<<edgeBlocks, 256, 0, stream>>>(edst, ebuf, menc, denom, NEDGES);
        k_edge_agg<<<NEDGES / 8, 256, 0, stream>>>(esrc, edst, ebuf, denom,
                                                   xalt, AGG, NEDGES);
        // x_next = sigma*sigmoid(agg) + (1-sigma)*x   (WMMA f32 + fused epilogue)
        // Writes over xalt (Wh no longer needed).
        k_highway<<<gemmBlocks, 256, 0, stream>>>(xcur, Whw, bhw, AGG, xalt, NNODES);
        float* t = xcur; xcur = xalt; xalt = t;
    }

    k_gather<<<(NBATCH * DFEAT + 255) / 256, 256, 0, stream>>>(xcur, rel,
                                                               bh, br, bt,
                                                               out, NBATCH);
}